// CausalSelfAttention_13211319402924
// MI455X (gfx1250) — compile-verified
//
#include <hip/hip_runtime.h>
#include <math.h>

// Causal self-attention for MI455X (gfx1250), fp32 end-to-end using
// V_WMMA_F32_16X16X4_F32. Workload is HBM-bound (~300MB traffic @ 23.3 TB/s
// dwarfs ~120 GFLOP of matrix math), so native fp32 WMMA costs nothing vs
// low-precision and matches the fp32 reference exactly.
//
// LDS layouts are k-pair interleaved ([k/2][n][2]) so every WMMA B-fragment
// is a single aligned ds_load_b64 into a contiguous even VGPR pair (no
// v_mov shuffles), with pad strides chosen so the two lane-halves (pair p
// and p+1) hit disjoint LDS bank sets.

typedef __attribute__((ext_vector_type(2))) float v2f;
typedef __attribute__((ext_vector_type(8))) float v8f;

#define BB   4
#define TT   2048
#define DD   1024
#define NH   16
#define HDM  64

static __device__ __forceinline__ v8f wmma_f32(v2f a, v2f b, v8f c) {
  return __builtin_amdgcn_wmma_f32_16x16x4_f32(false, a, false, b, (short)0, c,
                                               false, false);
}

// ---------------------------------------------------------------------------
// Kernel 1: Y = X @ W^T + bias, written into [B, H, T, HD] layout.
// Block = 256 threads (8 wave32), tile = 64(M) x 64(N), BK = 16.
// ---------------------------------------------------------------------------
__global__ __launch_bounds__(256) void qkv_gemm_kernel(
    const float* __restrict__ X,     // [B*T, D]
    const float* __restrict__ W,     // [D, D], row j = output feature
    const float* __restrict__ bias,  // [D]
    float* __restrict__ Y)           // [B, H, T, HD]
{
  // A tile: [m][k], even row stride -> aligned b64 fragment loads.
  __shared__ __align__(16) float sA[64][18];
  // B tile, k-pair interleaved: sBp[k/2][j][2]; stride 160 floats per pair
  // row (160 mod 64 == 32 -> disjoint banks for the two lane-halves).
  __shared__ __align__(16) float sBp[8][80][2];

  const int tid  = threadIdx.x;
  const int lane = tid & 31;
  const int wave = tid >> 5;
  const int mg   = wave >> 1;     // 0..3 -> 16-row strip
  const int half = wave & 1;      // 0..1 -> 32-col strip
  const int ln   = lane & 15;
  const int kh   = lane >> 4;     // 0/1 -> K offset 0/2 in frags

  const int m0 = blockIdx.y * 64; // token-tile base
  const int j0 = blockIdx.x * 64; // feature-tile base

  v8f acc0 = {};
  v8f acc1 = {};

  const int row = tid >> 2;        // 0..63
  const int cg  = (tid & 3) * 4;   // 0,4,8,12

  for (int k0 = 0; k0 < DD; k0 += 16) {
    // Stage X tile and W tile (coalesced float4 per 4 threads/row).
    {
      const float4 xa =
          *(const float4*)(X + (size_t)(m0 + row) * DD + (k0 + cg));
      sA[row][cg + 0] = xa.x; sA[row][cg + 1] = xa.y;
      sA[row][cg + 2] = xa.z; sA[row][cg + 3] = xa.w;
      const float4 wa =
          *(const float4*)(W + (size_t)(j0 + row) * DD + (k0 + cg));
      *(float2*)&sBp[(cg >> 1) + 0][row][0] = make_float2(wa.x, wa.y);
      *(float2*)&sBp[(cg >> 1) + 1][row][0] = make_float2(wa.z, wa.w);
    }
    __syncthreads();

#pragma unroll
    for (int i = 0; i < 4; ++i) {
      const int kk = i * 4 + kh * 2;       // even
      const v2f a  = *(const v2f*)&sA[mg * 16 + ln][kk];
      const v2f b0 = *(const v2f*)&sBp[kk >> 1][half * 32 + ln][0];
      const v2f b1 = *(const v2f*)&sBp[kk >> 1][half * 32 + 16 + ln][0];
      acc0 = wmma_f32(a, b0, acc0);
      acc1 = wmma_f32(a, b1, acc1);
    }
    __syncthreads();
  }

  // Epilogue: add bias, scatter into [B,H,T,HD].
#pragma unroll
  for (int sub = 0; sub < 2; ++sub) {
    const int j  = j0 + half * 32 + sub * 16 + ln;
    const float bj = bias[j];
    const int h = j >> 6;
    const int d = j & 63;
#pragma unroll
    for (int r = 0; r < 8; ++r) {
      const int n  = m0 + mg * 16 + kh * 8 + r;  // global token index
      const int bi = n >> 11;                    // / T
      const int ti = n & (TT - 1);
      const float v = (sub ? acc1[r] : acc0[r]) + bj;
      Y[(((size_t)bi * NH + h) * TT + ti) * HDM + d] = v;
    }
  }
}

// ---------------------------------------------------------------------------
// Kernel 2: flash attention, causal, per (b, h, 64-query tile).
// Block = 256 threads (8 wave32): mg = wave>>1 -> 16 query rows,
// half = wave&1 -> key half (scores) / 32-wide value-dim half (output).
// KV chunk = 32 keys. Online softmax stats kept in LDS.
// ---------------------------------------------------------------------------
__global__ __launch_bounds__(256) void flash_attn_kernel(
    const float* __restrict__ Q,    // [B,H,T,HD]
    const float* __restrict__ K,    // [B,H,T,HD]
    const float* __restrict__ V,    // [B,H,T,HD]
    const float* __restrict__ mask, // [B,1,1,T] -> flat [B*T]
    float* __restrict__ O)          // [B,T,D]
{
  // K chunk transposed + d-pair interleaved: sKTp[d/2][kk][2], stride 96.
  __shared__ __align__(16) float sKTp[32][48][2];
  // V chunk, kk-pair interleaved: sVp[kk/2][d][2], stride 160.
  __shared__ __align__(16) float sVp[16][80][2];
  // Scores/probs [q][kk]; even row stride for aligned b64 A-fragment loads.
  __shared__ __align__(16) float sS[64][34];
  __shared__ float sM[64], sL[64], sC[64];  // row max / sum / correction

  const int tid  = threadIdx.x;
  const int lane = tid & 31;
  const int wave = tid >> 5;
  const int mg   = wave >> 1;
  const int half = wave & 1;
  const int ln   = lane & 15;
  const int kh   = lane >> 4;

  const int q0 = blockIdx.x * 64;
  const int h  = blockIdx.y;
  const int b  = blockIdx.z;

  const size_t bh = ((size_t)b * NH + h) * TT;
  const float* Qp = Q + bh * HDM;
  const float* Kp = K + bh * HDM;
  const float* Vp = V + bh * HDM;

  // Preload Q fragments for this wave's 16 query rows (d = 0..63).
  v2f aq[16];
  {
    const float* qr = Qp + (size_t)(q0 + mg * 16 + ln) * HDM;
#pragma unroll
    for (int i = 0; i < 16; ++i) {
      const float2 qf = *(const float2*)(qr + i * 4 + kh * 2);
      aq[i].x = qf.x;
      aq[i].y = qf.y;
    }
  }

  if (tid < 64) { sM[tid] = -INFINITY; sL[tid] = 0.0f; }

  v8f o0 = {};
  v8f o1 = {};

  const int nchunks = (q0 + 64) >> 5;  // causal: keys 0 .. q0+63
  const int krow = tid >> 3;           // 0..31: key row staged by this thread
  const int dg   = (tid & 7) * 8;      // 8 floats per thread per row
  const float scl = 0.125f;            // 1/sqrt(HD)

  for (int c = 0; c < nchunks; ++c) {
    const int kc = c * 32;

    // Stage K (transposed, d-pair interleaved) and V (kk-pair interleaved).
    {
      const float* kr = Kp + (size_t)(kc + krow) * HDM + dg;
      const float* vr = Vp + (size_t)(kc + krow) * HDM + dg;
      const float4 ka = *(const float4*)(kr);
      const float4 kb = *(const float4*)(kr + 4);
      *(float2*)&sKTp[(dg >> 1) + 0][krow][0] = make_float2(ka.x, ka.y);
      *(float2*)&sKTp[(dg >> 1) + 1][krow][0] = make_float2(ka.z, ka.w);
      *(float2*)&sKTp[(dg >> 1) + 2][krow][0] = make_float2(kb.x, kb.y);
      *(float2*)&sKTp[(dg >> 1) + 3][krow][0] = make_float2(kb.z, kb.w);
      const float4 va = *(const float4*)(vr);
      const float4 vb = *(const float4*)(vr + 4);
      const int kp = krow >> 1, ks = krow & 1;
      sVp[kp][dg + 0][ks] = va.x; sVp[kp][dg + 1][ks] = va.y;
      sVp[kp][dg + 2][ks] = va.z; sVp[kp][dg + 3][ks] = va.w;
      sVp[kp][dg + 4][ks] = vb.x; sVp[kp][dg + 5][ks] = vb.y;
      sVp[kp][dg + 6][ks] = vb.z; sVp[kp][dg + 7][ks] = vb.w;
      if (c + 1 < nchunks) {  // prefetch next chunk (global_prefetch_b8)
        __builtin_prefetch(kr + 32 * HDM, 0, 1);
        __builtin_prefetch(vr + 32 * HDM, 0, 1);
      }
    }
    __syncthreads();

    // S = Q @ K^T for this wave's 16q x 16k sub-tile (K-dim = HD = 64).
    v8f s = {};
#pragma unroll
    for (int i = 0; i < 16; ++i) {
      const int dd = i * 4 + kh * 2;  // even
      const v2f bf = *(const v2f*)&sKTp[dd >> 1][half * 16 + ln][0];
      s = wmma_f32(aq[i], bf, s);
    }

    // Scale, add padding mask, apply causal mask, write to LDS.
    {
      const int kkl = half * 16 + ln;
      const int kg  = kc + kkl;
      const float am = mask[(size_t)b * TT + kg];
#pragma unroll
      for (int r = 0; r < 8; ++r) {
        const int qg = q0 + mg * 16 + kh * 8 + r;
        float sv = s[r] * scl + am;
        if (kg > qg) sv = -INFINITY;
        sS[mg * 16 + kh * 8 + r][kkl] = sv;
      }
    }
    __syncthreads();

    // Online softmax statistics: one thread per query row.
    if (tid < 64) {
      const float mo = sM[tid];
      float rmax = mo;
#pragma unroll 8
      for (int k = 0; k < 32; ++k) rmax = fmaxf(rmax, sS[tid][k]);
      const float corr = (mo == -INFINITY) ? 0.0f : __expf(mo - rmax);
      float lsum = 0.0f;
#pragma unroll 8
      for (int k = 0; k < 32; ++k) {
        const float sv = sS[tid][k];
        const float p = (sv == -INFINITY) ? 0.0f : __expf(sv - rmax);
        sS[tid][k] = p;
        lsum += p;
      }
      sL[tid] = sL[tid] * corr + lsum;
      sM[tid] = rmax;
      sC[tid] = corr;
    }
    __syncthreads();

    // Rescale running output, then O += P @ V (K-dim = 32 keys).
#pragma unroll
    for (int r = 0; r < 8; ++r) {
      const float cr = sC[mg * 16 + kh * 8 + r];
      o0[r] *= cr;
      o1[r] *= cr;
    }
#pragma unroll
    for (int i = 0; i < 8; ++i) {
      const int kk = i * 4 + kh * 2;  // even
      const v2f ap  = *(const v2f*)&sS[mg * 16 + ln][kk];
      const v2f bv0 = *(const v2f*)&sVp[kk >> 1][half * 32 + ln][0];
      const v2f bv1 = *(const v2f*)&sVp[kk >> 1][half * 32 + 16 + ln][0];
      o0 = wmma_f32(ap, bv0, o0);
      o1 = wmma_f32(ap, bv1, o1);
    }
    __syncthreads();
  }

  // Normalize and store: merge heads into [B, T, D].
#pragma unroll
  for (int r = 0; r < 8; ++r) {
    const int ql  = mg * 16 + kh * 8 + r;
    const float inv = 1.0f / sL[ql];
    const int t  = q0 + ql;
    const int d0 = half * 32 + ln;
    float* op = O + ((size_t)b * TT + t) * DD + h * HDM;
    op[d0]      = o0[r] * inv;
    op[d0 + 16] = o1[r] * inv;
  }
}

// ---------------------------------------------------------------------------
extern "C" void kernel_launch(void* const* d_in, const int* in_sizes, int n_in,
                              void* d_out, int out_size, void* d_ws,
                              size_t ws_size, hipStream_t stream) {
  (void)in_sizes; (void)n_in; (void)out_size; (void)ws_size;
  const float* X    = (const float*)d_in[0];  // [B,T,D]
  const float* mask = (const float*)d_in[1];  // [B,1,1,T]
  const float* Wq   = (const float*)d_in[2];
  const float* bq   = (const float*)d_in[3];
  const float* Wk   = (const float*)d_in[4];
  const float* bk   = (const float*)d_in[5];
  const float* Wv   = (const float*)d_in[6];
  const float* bv   = (const float*)d_in[7];
  float* out = (float*)d_out;
  float* ws  = (float*)d_ws;

  const size_t QS = (size_t)BB * NH * TT * HDM;  // 8.4M floats (32 MB)
  float* Qb = ws;
  float* Kb = ws + QS;
  float* Vb = ws + 2 * QS;

  dim3 gg(DD / 64, (BB * TT) / 64);  // 16 x 128
  qkv_gemm_kernel<<<gg, 256, 0, stream>>>(X, Wq, bq, Qb);
  qkv_gemm_kernel<<<gg, 256, 0, stream>>>(X, Wk, bk, Kb);
  qkv_gemm_kernel<<<gg, 256, 0, stream>>>(X, Wv, bv, Vb);

  dim3 ga(TT / 64, NH, BB);  // 32 x 16 x 4
  flash_attn_kernel<<<ga, 256, 0, stream>>>(Qb, Kb, Vb, mask, out);
}